// LlamaAttention_10307921510857
// MI455X (gfx1250) — compile-verified
//
#include <hip/hip_runtime.h>
#include <math.h>

// ---------------------------------------------------------------------------
// LlamaAttention decode-extend for MI455X (gfx1250), fp32 end-to-end.
// B=4, Q=4, HID=4096, NH=32, NKV=8, HD=128, GROUPS=4, CACHE_LEN=4096.
// Memory-bound (~300MB reads @ 23.3TB/s); all GEMMs via V_WMMA_F32_16X16X4_F32,
// operands streamed with global_load_b128 via a slot-permuted K ordering.
// ---------------------------------------------------------------------------

#define HID       4096
#define CACHE_LEN 4096
#define CACHE_MAX 4608
#define QKV_N     6144      // 4096 q + 1024 k + 1024 v
#define KEYS_PAD  4112      // 257 key-tiles * 16
#define SCALE     0.08838834764831845f   // 1/sqrt(128)

typedef float v2f __attribute__((ext_vector_type(2)));
typedef float v4f __attribute__((ext_vector_type(4)));
typedef float v8f __attribute__((ext_vector_type(8)));

// D(16x16,f32) = A(16x4,f32) * B(4x16,f32) + C
// A frag: lanes 0-15 -> M=lane ; lanes 16-31 -> M=lane-16 (slots 2hi, 2hi+1)
// B frag: lanes 0-15 -> N=lane ; lanes 16-31 -> N=lane-16 (slots 2hi, 2hi+1)
// C/D   : lanes 0-15 -> N=lane, VGPR g -> M=g ; lanes 16-31 -> M=8+g
__device__ inline v8f wmma_f32_k4(v2f a, v2f b, v8f c) {
  return __builtin_amdgcn_wmma_f32_16x16x4_f32(
      false, a, false, b, (short)0, c, false, false);
}

// Slot-permuted 8-wide K block: each lane holds contiguous float4 at
// (k0 + 4*hi); two WMMA steps consume .xy then .zw.  A and B use the same
// permutation, so the K-reduction is exact (order-permuted only).
__device__ inline v8f wmma_k8(v4f a4, v4f b4, v8f c) {
  c = wmma_f32_k4(a4.xy, b4.xy, c);
  c = wmma_f32_k4(a4.zw, b4.zw, c);
  return c;
}

// ---------------------------------------------------------------------------
// Generic 16xN = 16xK @ (NxK)^T GEMM.  One 16-wide N tile per block; the 4
// waves of the block each take K/4, then reduce partial C tiles through LDS.
// Weight source switches at feature boundaries n1/n2 (for fused QKV).
// ---------------------------------------------------------------------------
__global__ __launch_bounds__(128) void gemm16_kernel(
    const float* __restrict__ A, int K,
    const float* __restrict__ W0, const float* __restrict__ W1,
    const float* __restrict__ W2, int n1, int n2,
    float* __restrict__ out, int ldo)
{
  __shared__ float red[4][256];
  const int lane = threadIdx.x & 31;
  const int wave = threadIdx.x >> 5;
  const int n0   = blockIdx.x * 16;
  const int m    = lane & 15;
  const int hi   = lane >> 4;

  const float* W; int nb;
  if      (n0 >= n2) { W = W2; nb = n0 - n2; }
  else if (n0 >= n1) { W = W1; nb = n0 - n1; }
  else               { W = W0; nb = n0; }
  const float* __restrict__ wrow = W + (size_t)(nb + m) * K + 4 * hi;
  const float* __restrict__ arow = A + (size_t)m * K + 4 * hi;

  const int chunk = K >> 2;
  const int kbeg  = wave * chunk;
  const int kend  = kbeg + chunk;

  v8f c = {};
  for (int k = kbeg; k < kend; k += 32) {
    __builtin_prefetch(wrow + k + 512, 0, 1);   // global_prefetch_b8
#pragma unroll
    for (int kk = 0; kk < 32; kk += 8) {
      v4f a4 = *(const v4f*)(arow + k + kk);    // global_load_b128
      v4f b4 = *(const v4f*)(wrow + k + kk);    // global_load_b128
      c = wmma_k8(a4, b4, c);
    }
  }
#pragma unroll
  for (int g = 0; g < 8; ++g)
    red[wave][(g + 8*hi) * 16 + m] = c[g];
  __syncthreads();
  for (int e = threadIdx.x; e < 256; e += 128) {
    float s = red[0][e] + red[1][e] + red[2][e] + red[3][e];
    out[(size_t)(e >> 4) * ldo + n0 + (e & 15)] = s;
  }
}

// ---------------------------------------------------------------------------
// RoPE: qkv[16][6144] -> q_r[B][NKV][16][128] (rows = group*4 + s),
//                        k_r[B][NKV][4][128], v_n[B][NKV][4][128]
// ---------------------------------------------------------------------------
__global__ __launch_bounds__(256) void rope_kernel(
    const float* __restrict__ qkv, const float* __restrict__ cosb,
    const float* __restrict__ sinb, float* __restrict__ q_r,
    float* __restrict__ k_r, float* __restrict__ v_n)
{
  int idx = blockIdx.x * 256 + threadIdx.x;
  if (idx < 65536) {                      // Q heads
    int d = idx & 127, r = (idx >> 7) & 15, kvh = (idx >> 11) & 7, b = idx >> 14;
    int g = r >> 2, s = r & 3;
    int t = b * 4 + s, h = kvh * 4 + g;
    const float* base = qkv + (size_t)t * QKV_N + h * 128;
    float v  = base[d];
    float p  = base[(d < 64) ? d + 64 : d - 64];
    float rh = (d < 64) ? -p : p;
    int cs = t * 128 + d;
    q_r[idx] = v * cosb[cs] + rh * sinb[cs];
  } else if (idx < 81920) {               // K heads (new tokens)
    int i = idx - 65536;
    int d = i & 127, s = (i >> 7) & 3, kvh = (i >> 9) & 7, b = i >> 12;
    int t = b * 4 + s;
    const float* base = qkv + (size_t)t * QKV_N + 4096 + kvh * 128;
    float v  = base[d];
    float p  = base[(d < 64) ? d + 64 : d - 64];
    float rh = (d < 64) ? -p : p;
    int cs = t * 128 + d;
    k_r[i] = v * cosb[cs] + rh * sinb[cs];
  } else if (idx < 98304) {               // V heads (new tokens, no rope)
    int i = idx - 81920;
    int d = i & 127, s = (i >> 7) & 3, kvh = (i >> 9) & 7, b = i >> 12;
    int t = b * 4 + s;
    v_n[i] = qkv[(size_t)t * QKV_N + 5120 + kvh * 128 + d];
  }
}

// ---------------------------------------------------------------------------
// Scores: per (b, kvh, key-tile) one wave computes a 16x16 score tile
// over HD=128 (16 b128 loads + 32 fp32 WMMA).  Tile 256 holds the 4 new
// keys with the causal mask; pad columns get -inf.
// ---------------------------------------------------------------------------
__global__ __launch_bounds__(32) void scores_kernel(
    const float* __restrict__ q_r, const float* __restrict__ K_Cache,
    const float* __restrict__ k_r, float* __restrict__ scores)
{
  int bid = blockIdx.x;
  int kt  = bid % 257;
  int kvh = (bid / 257) & 7;
  int b   = bid / (257 * 8);
  int lane = threadIdx.x;
  int m = lane & 15, hi = lane >> 4;

  const float* __restrict__ qrow =
      q_r + ((size_t)((b * 8 + kvh) * 16 + m)) * 128 + 4 * hi;

  int key = kt * 16 + m;                 // B column: key index
  const float* krow; bool valid = true;
  if (kt < 256) {
    krow = K_Cache + ((size_t)(b * 8 + kvh) * CACHE_MAX + key) * 128 + 4 * hi;
  } else {
    int sn = key - CACHE_LEN;
    if (sn < 4) krow = k_r + ((size_t)((b * 8 + kvh) * 4 + sn)) * 128 + 4 * hi;
    else { valid = false; krow = k_r; }
  }

  v8f c = {};
#pragma unroll
  for (int k = 0; k < 128; k += 8) {
    v4f a4 = *(const v4f*)(qrow + k);
    v4f b4 = valid ? *(const v4f*)(krow + k) : (v4f){0.f, 0.f, 0.f, 0.f};
    c = wmma_k8(a4, b4, c);
  }

#pragma unroll
  for (int g = 0; g < 8; ++g) {
    int row = g + 8*hi;                  // query row (group*4 + s)
    int kc  = kt * 16 + m;
    float sc = c[g] * SCALE;
    if (kt == 256) {
      int sn = kc - CACHE_LEN, s = row & 3;
      if (sn >= 4 || sn > s) sc = -__builtin_inff();   // causal / pad mask
    }
    scores[((size_t)((b * 8 + kvh) * 16 + row)) * KEYS_PAD + kc] = sc;
  }
}

// ---------------------------------------------------------------------------
// Row softmax over 4112 entries (masked = -inf -> 0), in place.
// ---------------------------------------------------------------------------
__global__ __launch_bounds__(128) void softmax_kernel(float* __restrict__ scores)
{
  __shared__ float sred[128];
  float* row = scores + (size_t)blockIdx.x * KEYS_PAD;
  int tid = threadIdx.x;

  float mx = -3.4e38f;
  for (int i = tid; i < KEYS_PAD; i += 128) mx = fmaxf(mx, row[i]);
  sred[tid] = mx; __syncthreads();
  for (int s = 64; s > 0; s >>= 1) {
    if (tid < s) sred[tid] = fmaxf(sred[tid], sred[tid + s]);
    __syncthreads();
  }
  mx = sred[0]; __syncthreads();

  float sum = 0.0f;
  for (int i = tid; i < KEYS_PAD; i += 128) {
    float e = __expf(row[i] - mx);
    row[i] = e; sum += e;
  }
  sred[tid] = sum; __syncthreads();
  for (int s = 64; s > 0; s >>= 1) {
    if (tid < s) sred[tid] += sred[tid + s];
    __syncthreads();
  }
  float inv = 1.0f / sred[0];
  for (int i = tid; i < KEYS_PAD; i += 128) row[i] *= inv;
}

// ---------------------------------------------------------------------------
// attn @ V: per (b, kvh, 16-wide d-tile, k-split) one wave accumulates a
// 16x16 block of the 16x128 output.  Main loop: 1024-key aligned chunk with
// b128 probability loads; k-split 3 additionally handles the 4 new tokens.
// Partials go to 4 separate buffers (deterministic; reduced below).
// ---------------------------------------------------------------------------
__global__ __launch_bounds__(32) void av_kernel(
    const float* __restrict__ scores, const float* __restrict__ V_Cache,
    const float* __restrict__ v_n, float* __restrict__ partial)
{
  int bid = blockIdx.x;
  int ks  = bid & 3;
  int nt  = (bid >> 2) & 7;
  int kvh = (bid >> 5) & 7;
  int b   = bid >> 8;
  int lane = threadIdx.x;
  int m = lane & 15, hi = lane >> 4;

  const float* __restrict__ prow =
      scores + ((size_t)((b * 8 + kvh) * 16 + m)) * KEYS_PAD;
  int d = nt * 16 + m;
  const float* __restrict__ vbase =
      V_Cache + ((size_t)(b * 8 + kvh) * CACHE_MAX) * 128 + d;
  const float* __restrict__ vnew =
      v_n + ((size_t)(b * 8 + kvh) * 4) * 128 + d;

  const int kbeg = ks * 1024;
  const int kend = kbeg + 1024;

  v8f c = {};
  for (int k = kbeg; k < kend; k += 8) {
    // A: contiguous probabilities, slot-permuted b128 load
    v4f a4 = *(const v4f*)(prow + k + 4 * hi);
    int r0 = k + 4 * hi;                 // V rows matching the permutation
    v2f b1, b2;
    b1.x = vbase[(size_t)(r0    ) * 128]; b1.y = vbase[(size_t)(r0 + 1) * 128];
    b2.x = vbase[(size_t)(r0 + 2) * 128]; b2.y = vbase[(size_t)(r0 + 3) * 128];
    c = wmma_f32_k4(a4.xy, b1, c);
    c = wmma_f32_k4(a4.zw, b2, c);
  }
  if (ks == 3) {                         // tail: 4 new tokens (keys 4096..4099)
    v2f a, bb;
    a.x  = prow[CACHE_LEN + 2*hi];       a.y  = prow[CACHE_LEN + 2*hi + 1];
    bb.x = vnew[(2*hi) * 128];           bb.y = vnew[(2*hi + 1) * 128];
    c = wmma_f32_k4(a, bb, c);
  }

#pragma unroll
  for (int g = 0; g < 8; ++g) {
    int row = g + 8*hi;
    int h = kvh * 4 + (row >> 2), s = row & 3, t = b * 4 + s;
    partial[(size_t)ks * 65536 + (size_t)t * 4096 + h * 128 + d] = c[g];
  }
}

__global__ __launch_bounds__(256) void reduce4_kernel(
    const float* __restrict__ partial, float* __restrict__ attn)
{
  int i = blockIdx.x * 256 + threadIdx.x;
  if (i < 65536)
    attn[i] = partial[i] + partial[i + 65536] +
              partial[i + 131072] + partial[i + 196608];
}

// ---------------------------------------------------------------------------
extern "C" void kernel_launch(void* const* d_in, const int* in_sizes, int n_in,
                              void* d_out, int out_size, void* d_ws, size_t ws_size,
                              hipStream_t stream) {
  const float* hidden = (const float*)d_in[0];
  const float* cosb   = (const float*)d_in[1];
  const float* sinb   = (const float*)d_in[2];
  const float* Kc     = (const float*)d_in[3];
  const float* Vc     = (const float*)d_in[4];
  const float* Wq     = (const float*)d_in[5];
  const float* Wk     = (const float*)d_in[6];
  const float* Wv     = (const float*)d_in[7];
  const float* Wo     = (const float*)d_in[8];
  float* out = (float*)d_out;

  float* ws      = (float*)d_ws;
  float* qkv     = ws;                    //  16*6144      =   98304
  float* q_r     = qkv + 98304;           //  4*8*16*128   =   65536
  float* k_r     = q_r + 65536;           //  4*8*4*128    =   16384
  float* v_n     = k_r + 16384;           //  4*8*4*128    =   16384
  float* scores  = v_n + 16384;           //  512*4112     = 2105344
  float* partial = scores + 2105344;      //  4*65536      =  262144
  float* attn    = partial + 262144;      //  16*4096      =   65536
                                          //  total ~10.3 MB

  // 1) fused QKV projection: [16 x 6144] = hidden[16 x 4096] @ {Wq|Wk|Wv}^T
  gemm16_kernel<<<QKV_N / 16, 128, 0, stream>>>(
      hidden, HID, Wq, Wk, Wv, 4096, 5120, qkv, QKV_N);
  // 2) RoPE + split into q_r / k_r / v_n (caches are inputs: never mutated)
  rope_kernel<<<384, 256, 0, stream>>>(qkv, cosb, sinb, q_r, k_r, v_n);
  // 3) scores = q @ K^T * scale, causal-masked, rows padded to 4112
  scores_kernel<<<4 * 8 * 257, 32, 0, stream>>>(q_r, Kc, k_r, scores);
  // 4) row softmax (512 rows)
  softmax_kernel<<<512, 128, 0, stream>>>(scores);
  // 5) attn @ V (4-way deterministic K-split) + reduce
  av_kernel<<<4 * 8 * 8 * 4, 32, 0, stream>>>(scores, Vc, v_n, partial);
  reduce4_kernel<<<256, 256, 0, stream>>>(partial, attn);
  // 6) output projection: out[16 x 4096] = attn @ Wo^T
  gemm16_kernel<<<HID / 16, 128, 0, stream>>>(
      attn, 4096, Wo, Wo, Wo, 1 << 30, 1 << 30, out, HID);
}